// MeshConv_36893769072935
// MI455X (gfx1250) — compile-verified
//
#include <hip/hip_runtime.h>

typedef float v2f __attribute__((ext_vector_type(2)));
typedef float v8f __attribute__((ext_vector_type(8)));

// ---------------------------------------------------------------------------
// Degree count (src out-degree), integer atomics -> deterministic.
// ---------------------------------------------------------------------------
__global__ __launch_bounds__(256) void deg_kernel(const int* __restrict__ edges,
                                                  int* __restrict__ deg, int nE) {
    int e = blockIdx.x * 256 + threadIdx.x;
    if (e < nE) {
        int s = edges[2 * e];
        __hip_atomic_fetch_add(&deg[s], 1, __ATOMIC_RELAXED, __HIP_MEMORY_SCOPE_AGENT);
    }
}

__global__ __launch_bounds__(256) void dinv_kernel(const int* __restrict__ deg,
                                                   float* __restrict__ dinv, int M) {
    int i = blockIdx.x * 256 + threadIdx.x;
    if (i < M) {
        int d = deg[i];
        dinv[i] = (d > 0) ? rsqrtf((float)d) : 0.0f;
    }
}

__global__ __launch_bounds__(256) void norm_kernel(const int* __restrict__ edges,
                                                   const float* __restrict__ dinv,
                                                   float* __restrict__ nrm, int nE) {
    int e = blockIdx.x * 256 + threadIdx.x;
    if (e < nE) {
        int s = edges[2 * e];
        int q = edges[2 * e + 1];
        nrm[e] = -dinv[s] * dinv[q];
    }
}

// ---------------------------------------------------------------------------
// WMMA f32 GEMM:  out[M x N] = A[M x K] @ W[K x N]  (+ T + bias, optional ReLU)
// One 16-row tile per wave, N/16 accumulators, K stepped by 4 using
// v_wmma_f32_16x16x4_f32.
//
// W is staged in LDS *pair-packed*: pack[(p*N + n)*2 + j] = W[2p + j][n]
// so a B fragment (two consecutive K rows for one column) is a single
// 8-byte-aligned ds_load_b64 straight into the VGPR pair WMMA consumes —
// no repacking movs in the inner loop.
//
// Fragment layouts per CDNA5 ISA 7.12.2:
//   A 16x4 : lane m = lane&15, VGPR j holds K = (lane>>4)*2 + j
//   B 4x16 : lane n = lane&15, VGPR j holds K = (lane>>4)*2 + j
//   C 16x16: lane n = lane&15, VGPR v holds M = (lane>>4)*8 + v
// ---------------------------------------------------------------------------
template <int K, int N, bool ADD_T, bool RELU>
__global__ __launch_bounds__(256) void gemm_wmma_kernel(
    const float* __restrict__ A, const float* __restrict__ W,
    const float* __restrict__ T, const float* __restrict__ bias,
    float* __restrict__ out, int M) {
    __shared__ float ldsW[K * N];  // pair-packed layout
    for (int i = threadIdx.x; i < K * N; i += 256) {
        int k = i / N, n = i - k * N;
        int p = k >> 1, j = k & 1;
        ldsW[(p * N + n) * 2 + j] = W[i];
    }
    __syncthreads();

    const int wave = threadIdx.x >> 5;
    const int lane = threadIdx.x & 31;
    const int half = lane >> 4;   // 0 or 1
    const int l16  = lane & 15;
    const int koff = half * 2;    // this lane's K-pair base within a 4-step

    const int m0 = blockIdx.x * 128 + wave * 16;
    if (m0 >= M) return;          // wave-uniform exit

    int arow = m0 + l16;
    if (arow >= M) arow = M - 1;  // clamp: corrupted rows are exactly the unstored ones
    const float* Arow = A + (size_t)arow * K;

    constexpr int NT = N / 16;
    v8f acc[NT];
#pragma unroll
    for (int nt = 0; nt < NT; ++nt)
#pragma unroll
        for (int j = 0; j < 8; ++j) acc[nt][j] = 0.0f;

    for (int k = 0; k < K; k += 4) {
        v2f a = *(const v2f*)(Arow + k + koff);     // 8B aligned global load
        const int p = (k >> 1) + half;              // K-pair index for this lane
#pragma unroll
        for (int nt = 0; nt < NT; ++nt) {
            v2f b = *(const v2f*)(&ldsW[(p * N + nt * 16 + l16) * 2]);  // ds_load_b64
            acc[nt] = __builtin_amdgcn_wmma_f32_16x16x4_f32(
                false, a, false, b, (short)0, acc[nt], false, false);
        }
    }

#pragma unroll
    for (int v = 0; v < 8; ++v) {
        int m = m0 + half * 8 + v;
        if (m < M) {
#pragma unroll
            for (int nt = 0; nt < NT; ++nt) {
                int n = nt * 16 + l16;
                float r = acc[nt][v];
                if (ADD_T) r += T[(size_t)m * N + n] + bias[n];
                if (RELU) r = fmaxf(r, 0.0f);
                out[(size_t)m * N + n] = r;
            }
        }
    }
}

// ---------------------------------------------------------------------------
// Edge scatter: T[dst,:] += norm_e * Y[src,:].
// One wave per 32-dim chunk of one edge; the wave index is forced scalar via
// readfirstlane so edges[2e], edges[2e+1], nrm[e] become uniform (scalar)
// loads instead of 32 redundant per-lane VMEM loads. Only the coalesced
// Y[src] gather and the hardware global_atomic_add_f32 remain per-lane.
// ---------------------------------------------------------------------------
template <int DIM>  // DIM in {32, 64}; chunks per edge = DIM/32
__global__ __launch_bounds__(256) void scatter_kernel(
    const int* __restrict__ edges, const float* __restrict__ nrm,
    const float* __restrict__ Y, float* __restrict__ T, int nE) {
    constexpr int CPE = DIM / 32;                      // chunks (waves) per edge
    int wave = __builtin_amdgcn_readfirstlane(threadIdx.x >> 5);  // 0..7, scalar
    int gw   = blockIdx.x * 8 + wave;                  // global wave id (scalar)
    int e    = gw / CPE;                               // edge id (scalar)
    int ch   = gw - e * CPE;                           // chunk in edge (scalar)
    if (e >= nE) return;                               // wave-uniform exit
    int   s  = edges[2 * e];                           // uniform -> scalar load
    int   q  = edges[2 * e + 1];
    float nf = nrm[e];
    int   d  = ch * 32 + (threadIdx.x & 31);
    float v  = nf * Y[(size_t)s * DIM + d];
    __hip_atomic_fetch_add(&T[(size_t)q * DIM + d], v, __ATOMIC_RELAXED,
                           __HIP_MEMORY_SCOPE_AGENT);
}

// ---------------------------------------------------------------------------
// Orchestration. Pipeline (projection-before-scatter form):
//   Y1 = verts @ W1_1                      (WMMA, 192->64)
//   T1 = scatter(norm * Y1[src])           (64-wide atomics)
//   X1 = relu(verts @ W0_1 + T1 + b1)      (WMMA fused epilogue)
//   Y2 = X1 @ W1_2                          (WMMA, 64->32)
//   T2 = scatter(norm * Y2[src])           (32-wide atomics)
//   out = X1 @ W0_2 + T2 + b2              (WMMA fused epilogue)
// ---------------------------------------------------------------------------
extern "C" void kernel_launch(void* const* d_in, const int* in_sizes, int n_in,
                              void* d_out, int out_size, void* d_ws, size_t ws_size,
                              hipStream_t stream) {
    (void)n_in; (void)out_size; (void)ws_size;
    const float* verts = (const float*)d_in[0];
    const int*   edges = (const int*)d_in[1];
    const float* W0_1  = (const float*)d_in[2];
    const float* W1_1  = (const float*)d_in[3];
    const float* b1    = (const float*)d_in[4];
    const float* W0_2  = (const float*)d_in[5];
    const float* W1_2  = (const float*)d_in[6];
    const float* b2    = (const float*)d_in[7];
    float* out = (float*)d_out;

    const int M  = in_sizes[0] / 192;   // 100000 nodes
    const int nE = in_sizes[1] / 2;     // 1600000 edges

    char* ws = (char*)d_ws;
    size_t off = 0;
    int*   deg  = (int*)(ws + off);   off += (size_t)M * 4;
    float* dinv = (float*)(ws + off); off += (size_t)M * 4;
    float* nrm  = (float*)(ws + off); off += (size_t)nE * 4;
    float* bufA = (float*)(ws + off); off += (size_t)M * 64 * 4;   // Y1 then X1
    float* bufB = (float*)(ws + off); off += (size_t)M * 64 * 4;   // T1 then Y2|T2

    float* Y1 = bufA;
    float* X1 = bufA;
    float* T1 = bufB;
    float* Y2 = bufB;
    float* T2 = bufB + (size_t)M * 32;

    // Zero-init accumulators (graph-capturable memset nodes).
    hipMemsetAsync(deg, 0, (size_t)M * 4, stream);
    hipMemsetAsync(T1, 0, (size_t)M * 64 * 4, stream);

    deg_kernel<<<(nE + 255) / 256, 256, 0, stream>>>(edges, deg, nE);
    dinv_kernel<<<(M + 255) / 256, 256, 0, stream>>>(deg, dinv, M);
    norm_kernel<<<(nE + 255) / 256, 256, 0, stream>>>(edges, dinv, nrm, nE);

    // ---- Layer 1 ----
    gemm_wmma_kernel<192, 64, false, false><<<(M + 127) / 128, 256, 0, stream>>>(
        verts, W1_1, nullptr, nullptr, Y1, M);
    {   // nE edges * 2 waves each, 8 waves per block
        unsigned blocks = (unsigned)(((size_t)nE * 2 + 7) / 8);
        scatter_kernel<64><<<blocks, 256, 0, stream>>>(edges, nrm, Y1, T1, nE);
    }
    gemm_wmma_kernel<192, 64, true, true><<<(M + 127) / 128, 256, 0, stream>>>(
        verts, W0_1, T1, b1, X1, M);

    // Reuse bufB for layer 2 (T1 dead after the fused GEMM above).
    hipMemsetAsync(bufB, 0, (size_t)M * 64 * 4, stream);

    // ---- Layer 2 ----
    gemm_wmma_kernel<64, 32, false, false><<<(M + 127) / 128, 256, 0, stream>>>(
        X1, W1_2, nullptr, nullptr, Y2, M);
    {   // nE edges * 1 wave each
        unsigned blocks = (unsigned)(((size_t)nE + 7) / 8);
        scatter_kernel<32><<<blocks, 256, 0, stream>>>(edges, nrm, Y2, T2, nE);
    }
    gemm_wmma_kernel<64, 32, true, false><<<(M + 127) / 128, 256, 0, stream>>>(
        X1, W0_2, T2, b2, out, M);
}